// RelGCNddd_1958505087051
// MI455X (gfx1250) — compile-verified
//
#include <hip/hip_runtime.h>

// ---------------------------------------------------------------------------
// GCN (3 layers) for gfx1250 / MI455X.
//   layer: H = A_in @ W ; Agg[t] += H[s]*dinv[t]*dinv[s] (incl self loops);
//          out = Agg + b ; relu fused into next GEMM's A-staging.
// GEMM: V_WMMA_F32_16X16X4_F32 (fp32, wave32). A tile staged in LDS
// (coalesced, relu+pad fused, conflict-free +2 stride); W pre-padded so the
// K-loop has zero guards; uniform full-tile store fast path (N % 16 == 0).
// Scatter: coalesced global_atomic_add_f32; 61MB feature mats live in 192MB L2.
// ---------------------------------------------------------------------------

typedef __attribute__((ext_vector_type(2))) float v2f;
typedef __attribute__((ext_vector_type(8))) float v8f;

__global__ __launch_bounds__(256) void k_deg_init(float* __restrict__ deg, int n) {
  int i = blockIdx.x * blockDim.x + threadIdx.x;
  if (i < n) deg[i] = 1.0f;  // self loop contributes 1
}

__global__ __launch_bounds__(256) void k_deg_accum(float* __restrict__ deg,
                                                   const int* __restrict__ rows, int E) {
  int e = blockIdx.x * blockDim.x + threadIdx.x;
  if (e < E) atomicAdd(&deg[rows[e]], 1.0f);
}

__global__ __launch_bounds__(256) void k_deg_rsqrt(float* __restrict__ deg, int n) {
  int i = blockIdx.x * blockDim.x + threadIdx.x;
  if (i < n) deg[i] = rsqrtf(deg[i]);  // deg >= 1 always
}

// out[i*ld + j] = (j < F) ? bias[j] : 0
__global__ __launch_bounds__(256) void k_init_bias(float* __restrict__ out,
                                                   const float* __restrict__ bias,
                                                   int n, int ld, int F) {
  int i = blockIdx.x * blockDim.x + threadIdx.x;
  if (i < n * ld) {
    int j = i % ld;
    out[i] = (j < F) ? bias[j] : 0.0f;
  }
}

// Zero-padded weight copy: Wp[k][n] (KPAD x NCP) from W[Kreal x Ncols].
__global__ __launch_bounds__(256) void k_pad_w(const float* __restrict__ W,
                                               float* __restrict__ Wp,
                                               int Kreal, int Kpad, int Ncols, int NCP) {
  int i = blockIdx.x * blockDim.x + threadIdx.x;
  if (i >= Kpad * NCP) return;
  int k = i / NCP, n = i - k * NCP;
  Wp[i] = (k < Kreal && n < Ncols) ? W[(size_t)k * Ncols + n] : 0.0f;
}

// C[Nrows x NCP] = A[Nrows x LDA] @ Wp[KPAD x NCP].
// Block = WAVES waves; one 16-row tile per block, one 16-col tile per wave.
// Fragment layouts per CDNA5 ISA 7.12.2 (f32):
//   A 16x4 : lane L (m=L&15, h=L>>4): v0=A[m][2h],  v1=A[m][2h+1]
//   B 4x16 : lane L (n=L&15, h=L>>4): v0=Wp[2h][n], v1=Wp[2h+1][n]
//   C 16x16: VGPR g, lane L: row = g + 8h, col = n
template <int LDA, int KPAD, int KREAL, int NCP, bool RELU, int WAVES>
__global__ __launch_bounds__(WAVES * 32) void k_gemm_wmma(
    const float* __restrict__ A, const float* __restrict__ Wp,
    float* __restrict__ C, int Nrows) {
  constexpr int LSTR = KPAD + 2;  // +2 floats: break 64-bank conflicts, keep b64 align
  __shared__ float As[16 * LSTR];

  const int tid = threadIdx.x;
  const int rowBase = blockIdx.x << 4;

  // Cooperative coalesced staging of the 16 x KPAD A-tile (relu + k-pad fused).
  for (int i = tid; i < 16 * KPAD; i += WAVES * 32) {
    int r = i / KPAD, k = i - r * KPAD;
    int gr = rowBase + r;
    if (gr >= Nrows) gr = Nrows - 1;  // harmless clamp; stores are guarded
    float v = (k < KREAL) ? A[(size_t)gr * LDA + k] : 0.0f;
    if (RELU) v = fmaxf(v, 0.0f);
    As[r * LSTR + k] = v;
  }
  __syncthreads();

  const int wave = tid >> 5;
  const int lane = tid & 31;
  const int m = lane & 15, hlf = lane >> 4, kb = hlf << 1;
  const int colBase = (blockIdx.y * WAVES + wave) << 4;
  if (colBase >= NCP) return;  // spare wave in last y-block

  const float* __restrict__ arow = &As[m * LSTR];
  const float* __restrict__ wcol = Wp + colBase + m;

  v8f acc = {};
#pragma unroll 4
  for (int k0 = 0; k0 < KPAD; k0 += 4) {
    const int ka = k0 + kb;
    v2f av; av.x = arow[ka];               av.y = arow[ka + 1];  // ds_load b64
    v2f bv; bv.x = wcol[(size_t)ka * NCP]; bv.y = wcol[(size_t)(ka + 1) * NCP];
    acc = __builtin_amdgcn_wmma_f32_16x16x4_f32(false, av, false, bv,
                                                (short)0, acc, false, false);
  }

  // Store: uniform (scalar-branch) full-tile fast path -> 8 straight
  // global_store_b32 with immediate row offsets; guarded path is cold.
  float* __restrict__ crow = C + (size_t)(rowBase + (hlf << 3)) * NCP + colBase + m;
  if (rowBase + 16 <= Nrows) {
#pragma unroll
    for (int g = 0; g < 8; ++g) crow[(size_t)g * NCP] = acc[g];
  } else {
#pragma unroll
    for (int g = 0; g < 8; ++g)
      if (rowBase + (hlf << 3) + g < Nrows) crow[(size_t)g * NCP] = acc[g];
  }
}

// One wave per edge (E real + N self loops); lanes stride the feature dim.
__global__ __launch_bounds__(256) void k_scatter(
    const float* __restrict__ H, float* __restrict__ Aout,
    const int* __restrict__ rows, const int* __restrict__ cols,
    const float* __restrict__ dinv, int E, int Nnodes, int ldH, int ldA, int F) {
  int w    = blockIdx.x * (blockDim.x >> 5) + (threadIdx.x >> 5);
  int lane = threadIdx.x & 31;
  int total = E + Nnodes;
  if (w >= total) return;
  int t, s;
  if (w < E) { t = rows[w]; s = cols[w]; }
  else       { t = w - E;   s = t; }
  float nrm = dinv[t] * dinv[s];
  const float* __restrict__ hs = H + (size_t)s * ldH;
  float* __restrict__ at = Aout + (size_t)t * ldA;
  for (int j = lane; j < F; j += 32)
    atomicAdd(&at[j], hs[j] * nrm);
}

extern "C" void kernel_launch(void* const* d_in, const int* in_sizes, int n_in,
                              void* d_out, int out_size, void* d_ws, size_t ws_size,
                              hipStream_t stream) {
  const float* x  = (const float*)d_in[0];
  const int*   ei = (const int*)d_in[1];   // edge_index int32, [2,E] flat
  // d_in[2] = edge_attr (unused by reference)
  const float* W1 = (const float*)d_in[3];
  const float* b1 = (const float*)d_in[4];
  const float* W2 = (const float*)d_in[5];
  const float* b2 = (const float*)d_in[6];
  const float* W3 = (const float*)d_in[7];
  const float* b3 = (const float*)d_in[8];

  const int F0 = 128;
  const int N  = in_sizes[0] / F0;   // 50000
  const int E  = in_sizes[1] / 2;    // 400000
  const int F1 = 300, F1P = 304;     // layer1 out, padded
  const int F2 = 50,  F2P = 64;      // layer2 out, padded (K 50 -> 52)
  const int F3 = 10,  F3P = 16;

  char* ws = (char*)d_ws;
  auto align256 = [](size_t v) { return (v + 255) & ~(size_t)255; };
  size_t o = 0;
  float* dinv = (float*)(ws + o); o += align256((size_t)N * 4);
  float* BH   = (float*)(ws + o); o += align256((size_t)N * F1P * 4);  // pre-agg H
  float* BA   = (float*)(ws + o); o += align256((size_t)N * F1P * 4);  // agg (+bias)
  float* Wp1  = (float*)(ws + o); o += align256((size_t)128 * F1P * 4);
  float* Wp2  = (float*)(ws + o); o += align256((size_t)304 * F2P * 4);
  float* Wp3  = (float*)(ws + o); o += align256((size_t)52  * F3P * 4);

  const int* rows = ei;       // targets
  const int* cols = ei + E;   // sources
  float* out = (float*)d_out;

  const dim3 b256(256), b128(128);
  const int rowTiles  = (N + 15) / 16;        // 3125
  const int totalEdge = E + N;
  const int edgeBlks  = (totalEdge + 7) / 8;  // 8 waves per 256-thread block

  // --- normalization: dinv = rsqrt(1 + in-degree(targets)) ---
  k_deg_init <<<dim3((N + 255) / 256), b256, 0, stream>>>(dinv, N);
  k_deg_accum<<<dim3((E + 255) / 256), b256, 0, stream>>>(dinv, rows, E);
  k_deg_rsqrt<<<dim3((N + 255) / 256), b256, 0, stream>>>(dinv, N);

  // --- pre-pad weights (tiny) ---
  k_pad_w<<<dim3((128 * F1P + 255) / 256), b256, 0, stream>>>(W1, Wp1, 128, 128, F1, F1P);
  k_pad_w<<<dim3((304 * F2P + 255) / 256), b256, 0, stream>>>(W2, Wp2, F1, 304, F2, F2P);
  k_pad_w<<<dim3((52  * F3P + 255) / 256), b256, 0, stream>>>(W3, Wp3, F2, 52,  F3, F3P);

  // --- layer 1: x[N,128] @ W1 -> BH ; agg -> BA (+b1) ---
  k_gemm_wmma<128, 128, 128, 304, false, 4>
      <<<dim3(rowTiles, (F1P / 16 + 3) / 4), b128, 0, stream>>>(x, Wp1, BH, N);
  k_init_bias<<<dim3((N * F1P + 255) / 256), b256, 0, stream>>>(BA, b1, N, F1P, F1);
  k_scatter<<<dim3(edgeBlks), b256, 0, stream>>>(BH, BA, rows, cols, dinv, E, N, F1P, F1P, F1);

  // --- layer 2: relu(BA)[N,300] @ W2 -> BH ; agg -> BA (+b2) ---
  k_gemm_wmma<304, 304, 300, 64, true, 4>
      <<<dim3(rowTiles, 1), b128, 0, stream>>>(BA, Wp2, BH, N);
  k_init_bias<<<dim3((N * F2P + 255) / 256), b256, 0, stream>>>(BA, b2, N, F2P, F2);
  k_scatter<<<dim3(edgeBlks), b256, 0, stream>>>(BH, BA, rows, cols, dinv, E, N, F2P, F2P, F2);

  // --- layer 3: relu(BA)[N,50] @ W3 -> BH ; agg -> d_out (+b3), no relu ---
  k_gemm_wmma<64, 52, 50, 16, true, 4>
      <<<dim3(rowTiles, 1), b128, 0, stream>>>(BA, Wp3, BH, N);
  k_init_bias<<<dim3((N * F3 + 255) / 256), b256, 0, stream>>>(out, b3, N, F3, F3);
  k_scatter<<<dim3(edgeBlks), b256, 0, stream>>>(BH, out, rows, cols, dinv, E, N, F3P, F3, F3);
}